// Gen2SingleInputReadout_19009525252169
// MI455X (gfx1250) — compile-verified
//
#include <hip/hip_runtime.h>
#include <cstdint>

// ---------------- problem constants ----------------
#define T_LEN 2048
#define B_SZ  8
#define IN_D  512
#define DV    512
#define DK    128
#define CHK   128
#define NCH   16          // T_LEN / CHK
#define EPSV  1e-8f

// ---------------- ws layout (floats) ----------------
#define V_OFF   ((size_t)0)                              // NCH*B*CHK*DV = 8388608
#define KT_OFF  ((size_t)8388608)                        // NCH*B*CHK*DK = 2097152
#define QT_OFF  ((size_t)10485760)                       // 2097152
#define AR_OFF  ((size_t)12582912)                       // 2097152 (raw alpha)
#define CPL_OFF ((size_t)14680064)                       // NCH*B*DK = 16384
#define S_OFF   ((size_t)14696448)                       // B*DK*DV = 524288

typedef float v2f __attribute__((ext_vector_type(2)));
typedef float v8f __attribute__((ext_vector_type(8)));

// fp32 WMMA: D(16x16) = A(16x4) * B(4x16) + C, exact fp32 (matches reference math)
__device__ __forceinline__ v8f wmma4(v2f a, v2f b, v8f c) {
    return __builtin_amdgcn_wmma_f32_16x16x4_f32(
        /*neg_a=*/false, a, /*neg_b=*/false, b,
        /*c_mod=*/(short)0, c, /*reuse_a=*/false, /*reuse_b=*/false);
}

// CDNA5 async global->LDS copy, 16B per lane, tracked by ASYNCcnt.
// lds_byte = byte offset into LDS (dynamic LDS base == 0 when no static LDS).
__device__ __forceinline__ void async_copy16(uint32_t lds_byte, const void* gptr) {
    uint64_t ga = (uint64_t)(uintptr_t)gptr;
    asm volatile("global_load_async_to_lds_b128 %0, %1, off"
                 :: "v"(lds_byte), "v"(ga) : "memory");
}
__device__ __forceinline__ void wait_async0() {
    asm volatile("s_wait_asynccnt 0" ::: "memory");
}

__device__ __forceinline__ v2f lds2(const float* p) {   // 8B-aligned pair load
    return *(const v2f*)p;
}

// =====================================================================
// Kernel 1: fused projection GEMM.
// out rows m = t*B+b (M=16384), cols: [0,512)=v(Wv), [512,640)=k(Wk),
// [640,768)=q(Wq), [768,896)=sigmoid alpha(Wa).  K = 512.
// Grid (128, 7) blocks of 256 threads; 128x128 output tile per WG.
// __launch_bounds__(256,1): full VGPR budget, no accumulator spills.
// =====================================================================
__global__ void __launch_bounds__(256, 1)
proj_kernel(const float* __restrict__ x,
            const float* __restrict__ Wv, const float* __restrict__ bv,
            const float* __restrict__ Wk, const float* __restrict__ bk,
            const float* __restrict__ Wq, const float* __restrict__ bq,
            const float* __restrict__ Wa, const float* __restrict__ ba,
            float* __restrict__ ws)
{
    extern __shared__ float smem[];
    const int RSP = 36;                       // padded row stride (floats): 144B, 16B-mult, conflict-free
    float* xs  = smem;                        // 128 x 36
    float* wsh = smem + 128 * RSP;            // 128 x 36
    const uint32_t XS_B = 0u, WS_B = (uint32_t)(128 * RSP * 4);

    const int tid  = threadIdx.x;
    const int wave = tid >> 5, lane = tid & 31;
    const int g = lane >> 4, l16 = lane & 15;
    const int m0 = blockIdx.x * 128;
    const int nt = blockIdx.y;                // 0..6 (column block of 128)

    const float* wptr; const float* bptr;
    if (nt < 4)       { wptr = Wv + (size_t)nt * 128 * IN_D; bptr = bv + nt * 128; }
    else if (nt == 4) { wptr = Wk; bptr = bk; }
    else if (nt == 5) { wptr = Wq; bptr = bq; }
    else              { wptr = Wa; bptr = ba; }

    v8f acc[8] = {};                          // 8 column tiles of 16x16 per wave

    for (int kb = 0; kb < IN_D; kb += 32) {
        // ---- async stage x(128x32) and W(128x32) tiles ----
        #pragma unroll
        for (int i = 0; i < 4; i++) {
            int s = tid + i * 256;            // 1024 16B segments per tile
            int row = s >> 3, c16 = s & 7;
            uint32_t lb = (uint32_t)(row * RSP * 4 + c16 * 16);
            async_copy16(XS_B + lb, x    + (size_t)(m0 + row) * IN_D + kb + c16 * 4);
            async_copy16(WS_B + lb, wptr + (size_t)row        * IN_D + kb + c16 * 4);
        }
        wait_async0();
        __syncthreads();

        // ---- 8 K-steps of WMMA; A frag shared across the 8 column tiles ----
        const float* xrow = xs + (wave * 16 + l16) * RSP;
        #pragma unroll 2
        for (int k4 = 0; k4 < 8; k4++) {
            v2f a = lds2(xrow + k4 * 4 + 2 * g);
            #pragma unroll
            for (int j = 0; j < 8; j++) {     // B[kk][n] = W[n][kk] -> contiguous in wsh
                v2f b = lds2(wsh + (j * 16 + l16) * RSP + k4 * 4 + 2 * g);
                acc[j] = wmma4(a, b, acc[j]);
            }
        }
        __syncthreads();                      // before restaging
    }

    // ---- epilogue: bias (+sigmoid for alpha), scatter to chunked layout ----
    #pragma unroll
    for (int j = 0; j < 8; j++) {
        float bias = bptr[j * 16 + l16];
        #pragma unroll
        for (int r = 0; r < 8; r++) {
            int m  = m0 + wave * 16 + r + 8 * g;
            int tg = m >> 3, b = m & 7;           // m = t*B + b, B=8
            int c  = tg >> 7, ti = tg & 127;      // chunk, t-in-chunk
            float val = acc[j][r] + bias;
            if (nt < 4) {
                ws[V_OFF + ((((size_t)c * B_SZ + b) * CHK + ti) * DV) + nt * 128 + j * 16 + l16] = val;
            } else {
                size_t idx = (((size_t)c * B_SZ + b) * CHK + ti) * DK + j * 16 + l16;
                if      (nt == 4) ws[KT_OFF + idx] = val;
                else if (nt == 5) ws[QT_OFF + idx] = val;
                else              ws[AR_OFF + idx] = 1.0f / (1.0f + expf(-val));
            }
        }
    }
}

// =====================================================================
// Kernel 2: per-(chunk,batch,n) sequential cumprod; rewrites k -> k~ = k/(cp+eps),
// q -> q~ = q*cp, and stores cp_last. 16384 independent chains.
// =====================================================================
__global__ void cumprod_kernel(float* __restrict__ ws)
{
    int id = blockIdx.x * blockDim.x + threadIdx.x;   // [0, 16*8*128)
    int n = id & 127, b = (id >> 7) & 7, c = id >> 10;
    size_t base = (((size_t)c * B_SZ + b) * CHK) * DK + n;
    float cp = 1.0f;
    for (int t = 0; t < CHK; t++) {
        size_t idx = base + (size_t)t * DK;
        float a = fmaxf(ws[AR_OFF + idx], EPSV);
        cp *= a;
        ws[KT_OFF + idx] = ws[KT_OFF + idx] / (cp + EPSV);
        ws[QT_OFF + idx] = ws[QT_OFF + idx] * cp;
    }
    ws[CPL_OFF + ((size_t)c * B_SZ + b) * DK + n] = cp;
}

__global__ void zeroS_kernel(float* __restrict__ ws)
{
    size_t id = (size_t)blockIdx.x * blockDim.x + threadIdx.x;
    if (id < (size_t)B_SZ * DK * DV) ws[S_OFF + id] = 0.0f;
}

// =====================================================================
// Kernel 3 (x16, host-serialized scan): one WG per (d-tile, batch).
//   A = q~ @ k~^T (causal mask)          [128x128, K=128]
//   y = mask(A) @ v + q~ @ S_prev^T      [128x128, K=128 + K=128]
//   S = cpl[n] * (S + k~^T @ v)          [128x128, K=128]
// S stored as [b][n][d] in ws; each WG owns a disjoint (b, d-slice).
// =====================================================================
__global__ void __launch_bounds__(256, 1)
chunk_kernel(float* __restrict__ ws, float* __restrict__ out, int c)
{
    extern __shared__ float smem[];
    const int RS = 132;                           // padded stride: 528B (16B-mult), conflict-free
    float* kt_s = smem;                           // 128 x 132
    float* qt_s = smem + 128 * RS;
    float* A_s  = smem + 2 * 128 * RS;
    float* v_s  = smem + 3 * 128 * RS;
    const uint32_t KT_B = 0u, QT_B = (uint32_t)(128 * RS * 4), VS_B = (uint32_t)(3 * 128 * RS * 4);

    const int tid  = threadIdx.x;
    const int wave = tid >> 5, lane = tid & 31;
    const int g = lane >> 4, l16 = lane & 15;
    const int dt = blockIdx.x, b = blockIdx.y;

    const float* ktg = ws + KT_OFF + (((size_t)c * B_SZ + b) * CHK) * DK;
    const float* qtg = ws + QT_OFF + (((size_t)c * B_SZ + b) * CHK) * DK;
    const float* vg  = ws + V_OFF  + (((size_t)c * B_SZ + b) * CHK) * DV + dt * 128;
    float*       Sg  = ws + S_OFF  + (size_t)b * DK * DV + dt * 128;     // [n*DV + dlocal]
    const float* cplg = ws + CPL_OFF + ((size_t)c * B_SZ + b) * DK;

    // ---- async stage k~, q~, v tiles (each 128x128 f32) ----
    #pragma unroll
    for (int i = 0; i < 16; i++) {
        int s = tid + i * 256;                    // 4096 16B segments per tile
        int row = s >> 5, c16 = s & 31;
        uint32_t lb = (uint32_t)(row * RS * 4 + c16 * 16);
        async_copy16(KT_B + lb, ktg + (size_t)row * DK + c16 * 4);
        async_copy16(QT_B + lb, qtg + (size_t)row * DK + c16 * 4);
        async_copy16(VS_B + lb, vg  + (size_t)row * DV + c16 * 4);
    }
    wait_async0();
    __syncthreads();

    // ---- step A: A[t,s] = sum_n q~[t,n] k~[s,n], causal mask, into LDS ----
    {
        const int t_base = wave * 16;
        const float* qrow = qt_s + (t_base + l16) * RS;
        for (int j = 0; j < 8; j++) {
            v8f acc = {};
            const float* krow = kt_s + (j * 16 + l16) * RS;   // B[kk=n][col=s] = kt[s][n]
            #pragma unroll 4
            for (int k4 = 0; k4 < 32; k4++) {
                v2f a  = lds2(qrow + k4 * 4 + 2 * g);
                v2f bf = lds2(krow + k4 * 4 + 2 * g);
                acc = wmma4(a, bf, acc);
            }
            #pragma unroll
            for (int r = 0; r < 8; r++) {
                int t = t_base + r + 8 * g;
                int s = j * 16 + l16;
                A_s[t * RS + s] = (s <= t) ? acc[r] : 0.0f;
            }
        }
    }
    __syncthreads();

    // ---- step y: y = A@v  +  q~ @ S_prev^T (S read from L2) ----
    {
        const int t_base = wave * 16;
        const float* arow = A_s  + (t_base + l16) * RS;
        const float* qrow = qt_s + (t_base + l16) * RS;
        for (int j = 0; j < 8; j++) {
            v8f acc = {};
            const int dcol = j * 16 + l16;
            #pragma unroll 4
            for (int k4 = 0; k4 < 32; k4++) {            // intra: K = s
                int kk = k4 * 4 + 2 * g;
                v2f a = lds2(arow + kk);
                v2f bf; bf.x = v_s[kk * RS + dcol]; bf.y = v_s[(kk + 1) * RS + dcol];
                acc = wmma4(a, bf, acc);
            }
            #pragma unroll 4
            for (int k4 = 0; k4 < 32; k4++) {            // inter: K = n
                int kk = k4 * 4 + 2 * g;
                v2f a = lds2(qrow + kk);
                v2f bf; bf.x = Sg[(size_t)kk * DV + dcol]; bf.y = Sg[(size_t)(kk + 1) * DV + dcol];
                acc = wmma4(a, bf, acc);
            }
            #pragma unroll
            for (int r = 0; r < 8; r++) {
                int t = t_base + r + 8 * g;
                out[(((size_t)c * CHK + t) * B_SZ + b) * DV + dt * 128 + dcol] = acc[r];
            }
        }
    }
    __syncthreads();    // all S_prev reads complete before S is overwritten

    // ---- step S: S[n,d] = cpl[n] * (S[n,d] + sum_s k~[s,n] v[s,d]) ----
    {
        const int n_base = wave * 16;
        for (int j = 0; j < 8; j++) {
            v8f acc = {};
            const int dcol = j * 16 + l16;
            #pragma unroll 4
            for (int k4 = 0; k4 < 32; k4++) {
                int kk = k4 * 4 + 2 * g;
                v2f a;  a.x  = kt_s[kk * RS + n_base + l16]; a.y  = kt_s[(kk + 1) * RS + n_base + l16];
                v2f bf; bf.x = v_s [kk * RS + dcol];         bf.y = v_s [(kk + 1) * RS + dcol];
                acc = wmma4(a, bf, acc);
            }
            #pragma unroll
            for (int r = 0; r < 8; r++) {
                int n = n_base + r + 8 * g;
                float cpl = cplg[n];
                size_t idx = (size_t)n * DV + dcol;
                Sg[idx] = cpl * (Sg[idx] + acc[r]);
            }
        }
    }
}

// =====================================================================
extern "C" void kernel_launch(void* const* d_in, const int* in_sizes, int n_in,
                              void* d_out, int out_size, void* d_ws, size_t ws_size,
                              hipStream_t stream) {
    (void)in_sizes; (void)n_in; (void)out_size; (void)ws_size;
    const float* x  = (const float*)d_in[0];
    const float* Wv = (const float*)d_in[1]; const float* bv = (const float*)d_in[2];
    const float* Wk = (const float*)d_in[3]; const float* bk = (const float*)d_in[4];
    const float* Wq = (const float*)d_in[5]; const float* bq = (const float*)d_in[6];
    const float* Wa = (const float*)d_in[7]; const float* ba = (const float*)d_in[8];
    float* out = (float*)d_out;
    float* ws  = (float*)d_ws;

    const size_t projLds = (size_t)2 * 128 * 36 * 4;    //  36 KB
    const size_t chLds   = (size_t)4 * 128 * 132 * 4;   // 264 KB (<= 320 KB/WGP)
    (void)hipFuncSetAttribute((const void*)chunk_kernel,
                              hipFuncAttributeMaxDynamicSharedMemorySize, (int)chLds);

    proj_kernel<<<dim3(128, 7), 256, projLds, stream>>>(x, Wv, bv, Wk, bk, Wq, bq, Wa, ba, ws);
    cumprod_kernel<<<64, 256, 0, stream>>>(ws);
    zeroS_kernel<<<2048, 256, 0, stream>>>(ws);
    for (int c = 0; c < NCH; c++)
        chunk_kernel<<<dim3(4, 8), 256, chLds, stream>>>(ws, out, c);
}